// DM_31988916421064
// MI455X (gfx1250) — compile-verified
//
#include <hip/hip_runtime.h>
#include <stdint.h>

// ---------------- tunables ----------------
#define RPB   128          // rows per block == threads per block (4 wave32)
#define TT    64           // time steps staged per tile
#define TT4   (TT/4)       // float4 chunks per row per tile = 16
#define CHUNKS (RPB*TT4)   // float4 chunks per tile = 2048 (32 KB)
#define DT_C  0.01f

#define AS1 __attribute__((address_space(1)))
#define AS3 __attribute__((address_space(3)))

typedef int   v4i __attribute__((__vector_size__(16)));
typedef float v4f __attribute__((__vector_size__(16)));

// cpol: gfx12+ encoding, TH in bits [2:0]; 1 = TH_LOAD_NT (stream once)
#define CPOL_NT 1

// ---- CDNA5 async global->LDS copy (ASYNCcnt path) ----
#if __has_builtin(__builtin_amdgcn_global_load_async_to_lds_b128)
__device__ __forceinline__ void async_copy_b128(const float4* g, float4* l) {
  __builtin_amdgcn_global_load_async_to_lds_b128((AS1 v4i*)g, (AS3 v4i*)l, 0, CPOL_NT);
}
#else
__device__ __forceinline__ void async_copy_b128(const float4* g, float4* l) {
  unsigned lds = (unsigned)(uintptr_t)(AS3 void*)l;
  asm volatile("global_load_async_to_lds_b128 %0, %1, off th:TH_LOAD_NT"
               :: "v"(lds), "v"((AS1 void*)g) : "memory");
}
#endif

#if __has_builtin(__builtin_amdgcn_s_wait_asynccnt)
#define WAIT_ASYNC(N) __builtin_amdgcn_s_wait_asynccnt(N)
#else
#define WAIT_ASYNC(N) asm volatile("s_wait_asynccnt " #N ::: "memory")
#endif

__global__ void __launch_bounds__(RPB)
dmp_scan_kernel(const float* __restrict__ force,  // (ROWS, T) contiguous
                const float* __restrict__ goal,   // (ROWS)
                const float* __restrict__ axp,    // scalar
                const float* __restrict__ bxp,    // scalar
                float* __restrict__ out,          // (ROWS, T)
                int T)
{
  __shared__ float4 sIn[2][CHUNKS];   // double-buffered force tiles (64 KB)
  __shared__ float4 sOut[CHUNKS];     // trajectory tile staging     (32 KB)

  const int tid = threadIdx.x;
  const int r0  = blockIdx.x * RPB;
  const int nTiles = T / TT;

  const float ax = axp[0];
  const float bx = bxp[0];
  const float g  = goal[r0 + tid];

  const float* fbase = force + (size_t)r0 * T;
  float*       obase = out   + (size_t)r0 * T;

  // ---- stage tile 0 (coalesced: 16 consecutive float4 per row) ----
  #pragma unroll
  for (int k = 0; k < TT4; ++k) {
    const int c   = k * RPB + tid;     // global chunk id within tile
    const int row = c >> 4;            // c / TT4
    const int t4  = c & (TT4 - 1);     // c % TT4
    const float4* src = (const float4*)(fbase + (size_t)row * T) + t4;
    float4* dst = &sIn[0][row * TT4 + (t4 ^ (row & (TT4 - 1)))]; // xor swizzle
    async_copy_b128(src, dst);
  }

  float x = 0.f, dx = 0.f;

  for (int tile = 0; tile < nTiles; ++tile) {
    const int buf = tile & 1;

    // ---- prefetch next tile into the other buffer, then drain current ----
    if (tile + 1 < nTiles) {
      const int nbuf = buf ^ 1;
      const size_t toff = (size_t)(tile + 1) * TT;
      #pragma unroll
      for (int k = 0; k < TT4; ++k) {
        const int c   = k * RPB + tid;
        const int row = c >> 4;
        const int t4  = c & (TT4 - 1);
        const float4* src = (const float4*)(fbase + (size_t)row * T + toff) + t4;
        float4* dst = &sIn[nbuf][row * TT4 + (t4 ^ (row & (TT4 - 1)))];
        async_copy_b128(src, dst);
      }
      WAIT_ASYNC(16);   // 16 newest (next tile) may remain; current tile done
    } else {
      WAIT_ASYNC(0);
    }
    __syncthreads();    // all waves' loads for `buf` visible; fences sOut reads

    // ---- serial scan over TT steps, exact reference arithmetic ----
    const int swz = tid & (TT4 - 1);
    #pragma unroll
    for (int i = 0; i < TT4; ++i) {
      const float4 f = sIn[buf][tid * TT4 + (i ^ swz)];
      float4 xs;
      { float ddx = ax * (bx * (g - x) - dx) + f.x; dx += ddx * DT_C; x += dx * DT_C; xs.x = x; }
      { float ddx = ax * (bx * (g - x) - dx) + f.y; dx += ddx * DT_C; x += dx * DT_C; xs.y = x; }
      { float ddx = ax * (bx * (g - x) - dx) + f.z; dx += ddx * DT_C; x += dx * DT_C; xs.z = x; }
      { float ddx = ax * (bx * (g - x) - dx) + f.w; dx += ddx * DT_C; x += dx * DT_C; xs.w = x; }
      sOut[tid * TT4 + (i ^ swz)] = xs;
    }
    __syncthreads();    // sOut tile complete

    // ---- coalesced non-temporal tile store (same transposed mapping) ----
    const size_t toff = (size_t)tile * TT;
    #pragma unroll
    for (int k = 0; k < TT4; ++k) {
      const int c   = k * RPB + tid;
      const int row = c >> 4;
      const int t4  = c & (TT4 - 1);
      const v4f v = *(const v4f*)&sOut[row * TT4 + (t4 ^ (row & (TT4 - 1)))];
      __builtin_nontemporal_store(v, (v4f*)(obase + (size_t)row * T + toff) + t4);
    }
    // top-of-loop WAIT+barrier of the next iteration orders these sOut reads
    // against the next compute phase's sOut writes, and the sOut-complete barrier orders compute reads
    // of sIn[buf] against the prefetch that overwrites it (distance 2).
  }
}

extern "C" void kernel_launch(void* const* d_in, const int* in_sizes, int n_in,
                              void* d_out, int out_size, void* d_ws, size_t ws_size,
                              hipStream_t stream) {
  (void)n_in; (void)d_ws; (void)ws_size; (void)out_size;
  const float* force = (const float*)d_in[0];  // (B, N, T) f32
  const float* goal  = (const float*)d_in[1];  // (B, N)    f32
  const float* axp   = (const float*)d_in[2];  // scalar f32 (device)
  const float* bxp   = (const float*)d_in[3];  // scalar f32 (device)
  float* out = (float*)d_out;

  const int rows = in_sizes[1];            // B*N = 32768
  const int T    = in_sizes[0] / rows;     // 4096
  const int grid = rows / RPB;             // 256 blocks of 128 threads

  dmp_scan_kernel<<<grid, RPB, 0, stream>>>(force, goal, axp, bxp, out, T);
}